// I_BPMLLLoss_68822555951606
// MI455X (gfx1250) — compile-verified
//
#include <hip/hip_runtime.h>

typedef float v2f __attribute__((ext_vector_type(2)));
typedef float v8f __attribute__((ext_vector_type(8)));

#define N_LABELS 1024
#define BATCH    32

// One workgroup: 8 waves. Wave w: row-tile t=w/4 (rows 16t..16t+15), K-quarter q=w%4.
// f32 WMMA A-matrix layout (16x4): lane L<16 -> (row L, K=0,1), lane L>=16 -> (row L-16, K=2,3).
// Each lane owns 2 partial-sum slots == 2 A-matrix cells of its row; B=ones reduces them per row.
__global__ __launch_bounds__(256)
void bpmll_wmma_kernel(const float* __restrict__ c,
                       const int*   __restrict__ y,
                       float*       __restrict__ out)
{
    __shared__ float partial[4][BATCH][6]; // [k-quarter][row][quantity]
    __shared__ float rowval[BATCH];

    const int tid  = threadIdx.x;
    const int wave = tid >> 5;        // 0..7
    const int lane = tid & 31;
    const int tile = wave >> 2;       // 0 or 1
    const int q    = wave & 3;        // K quarter (256 labels each)
    const int g    = lane >> 4;       // half-wave: selects A-matrix K pair {0,1} or {2,3}
    const int row  = tile * 16 + (lane & 15);

    const float* crow = c + row * (2 * N_LABELS); // interleaved c1/s1
    const int*   yrow = y + row * N_LABELS;

    // 6 quantities x 2 slots of f32 partial accumulators
    float sy0 = 0.f, sy1 = 0.f, syb0 = 0.f, syb1 = 0.f;
    float a0 = 0.f, a1 = 0.f, b0 = 0.f, b1 = 0.f;
    float c0s = 0.f, c1s = 0.f, d0 = 0.f, d1 = 0.f;

    // slot s of this lane covers labels k = 256*q + 4*i + 2*g + s  (i = 0..63)
    const int kbase = 256 * q + 2 * g;

#pragma unroll 4
    for (int i = 0; i < 64; ++i) {
        const int k0 = kbase + 4 * i;
        // float4 = { c1[k0], s1[k0], c1[k0+1], s1[k0+1] }, 16B aligned
        const float4 cv = *reinterpret_cast<const float4*>(crow + 2 * k0);
        const int2   yv = *reinterpret_cast<const int2*>(yrow + k0);

        { // slot 0
            const float yf = (float)yv.x, yb = 1.f - yf;
            const float e1 = __expf(-cv.x); // exp(-c1)
            const float e2 = __expf( cv.x); // exp(+c1)
            const float e3 = __expf(-cv.y); // exp(-s1)
            const float e4 = __expf( cv.y); // exp(+s1)
            sy0 += yf;        syb0 += yb;
            a0  += yf * e1;   b0   += yb * e2;
            c0s += yb * e3;   d0   += yf * e4;
        }
        { // slot 1
            const float yf = (float)yv.y, yb = 1.f - yf;
            const float e1 = __expf(-cv.z);
            const float e2 = __expf( cv.z);
            const float e3 = __expf(-cv.w);
            const float e4 = __expf( cv.w);
            sy1 += yf;        syb1 += yb;
            a1  += yf * e1;   b1   += yb * e2;
            c1s += yb * e3;   d1   += yf * e4;
        }
    }

    // Cross-lane reduction via full-precision f32 WMMA: D[m][*] = sum over the 4
    // partial columns of row m. B = all-ones 4x16, C = 0. EXEC is all ones here.
    v2f ones; ones[0] = 1.f; ones[1] = 1.f;
    v8f z = {};
    v2f m;
    m[0] = sy0;  m[1] = sy1;
    v8f Dsy  = __builtin_amdgcn_wmma_f32_16x16x4_f32(false, m, false, ones, (short)0, z, false, false);
    m[0] = syb0; m[1] = syb1;
    v8f Dsyb = __builtin_amdgcn_wmma_f32_16x16x4_f32(false, m, false, ones, (short)0, z, false, false);
    m[0] = a0;   m[1] = a1;
    v8f Da   = __builtin_amdgcn_wmma_f32_16x16x4_f32(false, m, false, ones, (short)0, z, false, false);
    m[0] = b0;   m[1] = b1;
    v8f Db   = __builtin_amdgcn_wmma_f32_16x16x4_f32(false, m, false, ones, (short)0, z, false, false);
    m[0] = c0s;  m[1] = c1s;
    v8f Dc   = __builtin_amdgcn_wmma_f32_16x16x4_f32(false, m, false, ones, (short)0, z, false, false);
    m[0] = d0;   m[1] = d1;
    v8f Dd   = __builtin_amdgcn_wmma_f32_16x16x4_f32(false, m, false, ones, (short)0, z, false, false);

    // C/D layout: VGPR j -> lane 0 holds row j, lane 16 holds row j+8 (any column).
    if ((lane & 15) == 0) {
        const int rbase = tile * 16 + g * 8;
#pragma unroll
        for (int j = 0; j < 8; ++j) {
            partial[q][rbase + j][0] = Dsy[j];
            partial[q][rbase + j][1] = Dsyb[j];
            partial[q][rbase + j][2] = Da[j];
            partial[q][rbase + j][3] = Db[j];
            partial[q][rbase + j][4] = Dc[j];
            partial[q][rbase + j][5] = Dd[j];
        }
    }
    __syncthreads();

    // Per-row loss: combine the 4 K-quarter partials, apply factored formula.
    if (tid < BATCH) {
        float sy = 0.f, syb = 0.f, a = 0.f, b = 0.f, cc = 0.f, d = 0.f;
#pragma unroll
        for (int qq = 0; qq < 4; ++qq) {
            sy  += partial[qq][tid][0];
            syb += partial[qq][tid][1];
            a   += partial[qq][tid][2];
            b   += partial[qq][tid][3];
            cc  += partial[qq][tid][4];
            d   += partial[qq][tid][5];
        }
        // BIAS=(1,1): denom = 2*yn*ybn + yn^2 + ybn^2
        const float denom = 2.f * sy * syb + sy * sy + syb * syb;
        // p1+p2+v1+v2 = (a+c)*(b+d)
        rowval[tid] = (a + cc) * (b + d) / denom;
    }
    __syncthreads();

    if (tid == 0) {
        float s = 0.f;
#pragma unroll
        for (int i = 0; i < BATCH; ++i) s += rowval[i];
        out[0] = s * (1.f / (float)BATCH);
    }
}

extern "C" void kernel_launch(void* const* d_in, const int* in_sizes, int n_in,
                              void* d_out, int out_size, void* d_ws, size_t ws_size,
                              hipStream_t stream) {
    const float* c = (const float*)d_in[0]; // (32, 2048) f32
    const int*   y = (const int*)d_in[1];   // (32, 1024) i32
    float* out = (float*)d_out;             // scalar f32
    (void)in_sizes; (void)n_in; (void)out_size; (void)d_ws; (void)ws_size;
    bpmll_wmma_kernel<<<1, 256, 0, stream>>>(c, y, out);
}